// GaussianSplattingModel_82480551952438
// MI455X (gfx1250) — compile-verified
//
#include <hip/hip_runtime.h>
#include <stdint.h>

// ---------------------------------------------------------------------------
// Gaussian-splatting forward (colors via SH3 + projected radii), MI455X.
// Memory-bound streaming kernel: one thread per gaussian, all 4 cameras.
// CDNA5 path: TDM (tensor_load_to_lds) DMAs each block's sh_rest tile
// (256 gaussians x 45 floats = 45 KB) into LDS; completion via TENSORcnt.
// rotations are provably unused: isotropic scale => cov3d = s^2 * I.
// ---------------------------------------------------------------------------

#define TILE_THREADS 256
#define SH_REST_FLOATS 45
#define TILE_ELEMS (TILE_THREADS * SH_REST_FLOATS)   // 11520 floats = 45 KB

typedef uint32_t u32x4 __attribute__((ext_vector_type(4)));
typedef int      i32x8 __attribute__((ext_vector_type(8)));
typedef int      i32x4 __attribute__((ext_vector_type(4)));

#if defined(__gfx1250__) && __has_builtin(__builtin_amdgcn_tensor_load_to_lds) && \
    __has_builtin(__builtin_amdgcn_s_wait_tensorcnt)
#define USE_TDM 1
#else
#define USE_TDM 0
#endif

__global__ __launch_bounds__(TILE_THREADS)
void gsplat_fwd_kernel(const float* __restrict__ means,
                       const float* __restrict__ scales,
                       const float* __restrict__ sh_dc,
                       const float* __restrict__ sh_rest,
                       const float* __restrict__ viewmats,  // (B,4,4) row-major
                       const float* __restrict__ campos,    // (B,3)
                       const int* __restrict__ p_h,
                       const int* __restrict__ p_w,
                       const int* __restrict__ p_f,
                       float* __restrict__ out,
                       int N, int B)
{
    __shared__ float sh_tile[TILE_ELEMS];
    const int tid = threadIdx.x;
    const int n   = blockIdx.x * TILE_THREADS + tid;

#if USE_TDM
    if (tid < 32) {  // wave 0 issues one TDM DMA for the whole block tile
        const uint64_t blockStart = (uint64_t)blockIdx.x * (uint64_t)TILE_ELEMS;
        const uint64_t gaddr  = (uint64_t)(uintptr_t)sh_rest + blockStart * 4ull;
        const uint32_t total  = (uint32_t)N * (uint32_t)SH_REST_FLOATS;   // 90M < 2^32
        const uint32_t td0    = total - (uint32_t)blockStart;             // remaining elems
        // low 32 bits of the flat LDS address == byte offset in the WG's LDS
        const uint32_t ldsoff = (uint32_t)(uintptr_t)&sh_tile[0];

        u32x4 g0;
        g0.x = 1u;                                   // count = 1 (valid descriptor)
        g0.y = ldsoff;                               // lds_addr
        g0.z = (uint32_t)gaddr;                      // global_addr[31:0]
        g0.w = (uint32_t)((gaddr >> 32) & 0x1FFFFFFull) | (2u << 30); // addr[56:32] | type=2

        i32x8 g1;
        g1.s0 = (int)(2u << 16);                     // workgroup_mask=0, data_size=2 (4B)
        g1.s1 = (int)((td0 & 0xFFFFu) << 16);        // tensor_dim0[15:0]  -> bits 63:48
        g1.s2 = (int)((td0 >> 16) | (1u << 16));     // tensor_dim0[31:16] | tensor_dim1=1
        g1.s3 = (int)((uint32_t)TILE_ELEMS << 16);   // tile_dim0 = 11520 (1D row)
        g1.s4 = 0;                                   // tile_dim1=0 (unused), tile_dim2=0
        g1.s5 = (int)TILE_ELEMS;                     // tensor_dim0_stride (single row)
        g1.s6 = 0;
        g1.s7 = 0;

        i32x4 gz4 = {0, 0, 0, 0};                    // groups 2/3 unused (<=2D tensor)
        i32x8 gz8 = {0, 0, 0, 0, 0, 0, 0, 0};        // extra group (clang-23 6-arg form)
        __builtin_amdgcn_tensor_load_to_lds(g0, g1, gz4, gz4, gz8, 0);
        __builtin_amdgcn_s_wait_tensorcnt(0);        // DMA complete before barrier
    }
    __syncthreads();
#else
    if (n < N) {
        const float* src = sh_rest + (size_t)n * SH_REST_FLOATS;
        float* dst = &sh_tile[tid * SH_REST_FLOATS];
        for (int i = 0; i < SH_REST_FLOATS; ++i) dst[i] = src[i];
    }
    __syncthreads();
#endif

    if (n >= N) return;

    // Pull my 45 SH-rest coefficients into VGPRs (stride 45 dwords: gcd(45,64)=1,
    // bank-conflict-free across the wave).
    float shr[SH_REST_FLOATS];
    {
        const float* myrow = &sh_tile[tid * SH_REST_FLOATS];
        #pragma unroll
        for (int i = 0; i < SH_REST_FLOATS; ++i) shr[i] = myrow[i];
    }

    const float mx = means[3 * (size_t)n + 0];
    const float my = means[3 * (size_t)n + 1];
    const float mz = means[3 * (size_t)n + 2];
    const float s  = scales[n];
    const float s2 = s * s;                      // cov3d = s^2 * I (isotropic, R orthonormal)
    const float dc0 = sh_dc[3 * (size_t)n + 0];
    const float dc1 = sh_dc[3 * (size_t)n + 1];
    const float dc2 = sh_dc[3 * (size_t)n + 2];

    const float fx = (float)(*p_f);              // fy == fx
    const float tanfovx = (float)(*p_w) / (2.0f * fx);
    const float tanfovy = (float)(*p_h) / (2.0f * fx);
    const float limx = 1.3f * tanfovx;
    const float limy = 1.3f * tanfovy;

    const size_t NN = (size_t)N;
    const size_t radii_base = (size_t)B * NN * 3;

    for (int b = 0; b < B; ++b) {
        const float* vm = viewmats + (size_t)b * 16;   // uniform -> scalar loads
        const float* cp = campos   + (size_t)b * 3;

        // ---- SH degree-3 color -------------------------------------------
        const float dx = mx - cp[0], dy = my - cp[1], dz = mz - cp[2];
        const float inv = rsqrtf(dx * dx + dy * dy + dz * dz);
        const float x = dx * inv, y = dy * inv, z = dz * inv;
        const float xx = x * x, yy = y * y, zz = z * z;
        const float xy = x * y, yz = y * z, xz = x * z;

        float w[16];
        w[0]  = 0.28209479177387814f;
        w[1]  = -0.4886025119029199f * y;
        w[2]  =  0.4886025119029199f * z;
        w[3]  = -0.4886025119029199f * x;
        w[4]  =  1.0925484305920792f * xy;
        w[5]  = -1.0925484305920792f * yz;
        w[6]  =  0.31539156525252005f * (2.0f * zz - xx - yy);
        w[7]  = -1.0925484305920792f * xz;
        w[8]  =  0.5462742152960396f * (xx - yy);
        w[9]  = -0.5900435899266435f * y * (3.0f * xx - yy);
        w[10] =  2.890611442640554f  * xy * z;
        w[11] = -0.4570457994644658f * y * (4.0f * zz - xx - yy);
        w[12] =  0.3731763325901154f * z * (2.0f * zz - 3.0f * xx - 3.0f * yy);
        w[13] = -0.4570457994644658f * x * (4.0f * zz - xx - yy);
        w[14] =  1.445305721320277f  * z * (xx - yy);
        w[15] = -0.5900435899266435f * x * (xx - yy - zz);

        float c0 = fmaf(w[0], dc0, 0.5f);
        float c1 = fmaf(w[0], dc1, 0.5f);
        float c2 = fmaf(w[0], dc2, 0.5f);
        #pragma unroll
        for (int k = 1; k < 16; ++k) {
            c0 = fmaf(w[k], shr[(k - 1) * 3 + 0], c0);
            c1 = fmaf(w[k], shr[(k - 1) * 3 + 1], c1);
            c2 = fmaf(w[k], shr[(k - 1) * 3 + 2], c2);
        }
        c0 = fmaxf(c0, 0.0f); c1 = fmaxf(c1, 0.0f); c2 = fmaxf(c2, 0.0f);

        // ---- projection + radius -----------------------------------------
        // t = mean @ vm[:3,:3] + vm[3,:3]
        const float t0 = fmaf(mx, vm[0], fmaf(my, vm[4], fmaf(mz, vm[8],  vm[12])));
        const float t1 = fmaf(mx, vm[1], fmaf(my, vm[5], fmaf(mz, vm[9],  vm[13])));
        const float t2 = fmaf(mx, vm[2], fmaf(my, vm[6], fmaf(mz, vm[10], vm[14])));
        const float tz = fmaxf(t2, 0.2f);
        const float iz = 1.0f / tz;
        const float txc = fminf(fmaxf(t0 * iz, -limx), limx) * tz;
        const float tyc = fminf(fmaxf(t1 * iz, -limy), limy) * tz;
        const float a0  = fx * iz;                   // J00 = J11
        const float a2x = -fx * txc * iz * iz;       // J02
        const float a2y = -fx * tyc * iz * iz;       // J12
        // T = J @ vm[:3,:3]^T  ->  T[r][k] = J[r][.] dot vm[k][.]
        const float T00 = fmaf(a0, vm[0], a2x * vm[2]);
        const float T01 = fmaf(a0, vm[4], a2x * vm[6]);
        const float T02 = fmaf(a0, vm[8], a2x * vm[10]);
        const float T10 = fmaf(a0, vm[1], a2y * vm[2]);
        const float T11 = fmaf(a0, vm[5], a2y * vm[6]);
        const float T12 = fmaf(a0, vm[9], a2y * vm[10]);
        // cov2d = s^2 * T T^T  (+0.3 diagonal)
        const float c00 = fmaf(s2, T00 * T00 + T01 * T01 + T02 * T02, 0.3f);
        const float c11 = fmaf(s2, T10 * T10 + T11 * T11 + T12 * T12, 0.3f);
        const float c01 = s2 * (T00 * T10 + T01 * T11 + T02 * T12);
        const float det = c00 * c11 - c01 * c01;
        const float mid = 0.5f * (c00 + c11);
        const float lam = mid + sqrtf(fmaxf(mid * mid - det, 0.1f));
        const float radii = 3.0f * sqrtf(lam);

        const size_t cb = ((size_t)b * NN + (size_t)n) * 3;
        out[cb + 0] = c0;
        out[cb + 1] = c1;
        out[cb + 2] = c2;
        out[radii_base + (size_t)b * NN + (size_t)n] = radii;
    }
}

extern "C" void kernel_launch(void* const* d_in, const int* in_sizes, int n_in,
                              void* d_out, int out_size, void* d_ws, size_t ws_size,
                              hipStream_t stream) {
    (void)n_in; (void)d_ws; (void)ws_size; (void)out_size;
    const float* means    = (const float*)d_in[0];
    const float* scales   = (const float*)d_in[1];
    // d_in[2] = rotations: provably unused (isotropic scale => cov3d = s^2 I)
    const float* sh_dc    = (const float*)d_in[3];
    const float* sh_rest  = (const float*)d_in[4];
    const float* viewmats = (const float*)d_in[5];
    const float* campos   = (const float*)d_in[6];
    const int*   p_h      = (const int*)d_in[7];
    const int*   p_w      = (const int*)d_in[8];
    const int*   p_f      = (const int*)d_in[9];
    float* out = (float*)d_out;

    const int N = in_sizes[0] / 3;      // means is (N,3)
    const int B = in_sizes[5] / 16;     // viewmats is (B,4,4)

    const int grid = (N + TILE_THREADS - 1) / TILE_THREADS;
    gsplat_fwd_kernel<<<grid, TILE_THREADS, 0, stream>>>(
        means, scales, sh_dc, sh_rest, viewmats, campos, p_h, p_w, p_f, out, N, B);
}